// CoulombEnergy_49563922596531
// MI455X (gfx1250) — compile-verified
//
#include <hip/hip_runtime.h>

typedef __attribute__((ext_vector_type(4))) float floatx4;
typedef __attribute__((ext_vector_type(4))) int   intx4;

#define GAS __attribute__((address_space(1)))
#define LAS __attribute__((address_space(3)))

#if defined(__HIP_DEVICE_COMPILE__) && \
    __has_builtin(__builtin_amdgcn_global_load_async_to_lds_b128) && \
    __has_builtin(__builtin_amdgcn_s_wait_asynccnt)
#define USE_ASYNC 1
#else
#define USE_ASYNC 0
#endif

// ---------------------------------------------------------------------------
// Zero the output buffer (harness poisons d_out with 0xAA before timing).
// ---------------------------------------------------------------------------
__global__ void coulomb_zero_f32(float* __restrict__ p, int n) {
  int i = blockIdx.x * blockDim.x + threadIdx.x;
  if (i < n) p[i] = 0.0f;
}

// ---------------------------------------------------------------------------
// Stage 1: voltage_atom[first[p]] += ecf * charges[second[p]] / dist[p]
// Streams 3 x 128-bit per lane through an async-to-LDS double-buffered
// pipeline (ASYNCcnt); gathers + atomics hit L2 (charges 4MB, voltage 4MB).
// ---------------------------------------------------------------------------
__global__ __launch_bounds__(256) void coulomb_pair_voltage_kernel(
    const float* __restrict__ charges,
    const float* __restrict__ dist,
    const int*   __restrict__ first,
    const int*   __restrict__ second,
    const float* __restrict__ ecf_p,
    float*       __restrict__ voltage,
    int n_pairs)
{
  const float ecf = *ecf_p;
  const int tid = threadIdx.x;
  const int n4  = n_pairs >> 2;

  // Scalar tail (n_pairs % 4): block 0 only, before any barrier.
  if (blockIdx.x == 0 && tid < (n_pairs & 3)) {
    int p = (n4 << 2) + tid;
    float v = ecf * charges[second[p]] / dist[p];
    atomicAdd(&voltage[first[p]], v);
  }
  if (n4 == 0) return;  // uniform across the whole grid

  const floatx4* d4 = (const floatx4*)dist;
  const intx4*   f4 = (const intx4*)first;
  const intx4*   s4 = (const intx4*)second;

#if USE_ASYNC
  __shared__ floatx4 s_dist[2][256];
  __shared__ intx4   s_first[2][256];
  __shared__ intx4   s_second[2][256];

  const int T = (n4 + 255) >> 8;  // tiles of 256 lanes * 4 pairs

  auto issue = [&](int t, int b) {
    int i = t * 256 + tid;
    if (i > n4 - 1) i = n4 - 1;  // clamp => every wave issues exactly 3 ops
    __builtin_amdgcn_global_load_async_to_lds_b128(
        (GAS intx4*)(d4 + i), (LAS intx4*)&s_dist[b][tid], 0, 0);
    __builtin_amdgcn_global_load_async_to_lds_b128(
        (GAS intx4*)(f4 + i), (LAS intx4*)&s_first[b][tid], 0, 0);
    __builtin_amdgcn_global_load_async_to_lds_b128(
        (GAS intx4*)(s4 + i), (LAS intx4*)&s_second[b][tid], 0, 0);
  };

  int t   = blockIdx.x;
  int buf = 0;
  if (t < T) issue(t, 0);
  for (; t < T; t += gridDim.x) {
    int  tn       = t + gridDim.x;
    bool has_next = tn < T;
    if (has_next) issue(tn, buf ^ 1);
    if (has_next) __builtin_amdgcn_s_wait_asynccnt(3);  // oldest 3 (tile t) done
    else          __builtin_amdgcn_s_wait_asynccnt(0);
    __syncthreads();  // tile t complete in LDS for all waves

    int i = t * 256 + tid;
    if (i < n4) {
      floatx4 dd = s_dist[buf][tid];
      intx4   ff = s_first[buf][tid];
      intx4   ss = s_second[buf][tid];
#pragma unroll
      for (int k = 0; k < 4; ++k) {
        float v = ecf * charges[ss[k]] / dd[k];
        atomicAdd(&voltage[ff[k]], v);
      }
    }
    __syncthreads();  // done reading buf before it is overwritten
    buf ^= 1;
  }
#else
  // Fallback: direct streaming with non-temporal hints (keep L2 for the
  // gather/atomic working set).
  const int stride = gridDim.x * 256;
  for (int i = blockIdx.x * 256 + tid; i < n4; i += stride) {
    floatx4 dd = __builtin_nontemporal_load(d4 + i);
    intx4   ff = __builtin_nontemporal_load(f4 + i);
    intx4   ss = __builtin_nontemporal_load(s4 + i);
#pragma unroll
    for (int k = 0; k < 4; ++k) {
      float v = ecf * charges[ss[k]] / dd[k];
      atomicAdd(&voltage[ff[k]], v);
    }
  }
#endif
}

// ---------------------------------------------------------------------------
// Stage 2: out_mol[mol[i]] += 0.5 * voltage[i] * charges[i]
// 1M atomics onto 10K slots; direct L2 atomics are cheaper than any
// LDS-privatized scheme at this block count.
// ---------------------------------------------------------------------------
__global__ void coulomb_atom_molecule_kernel(
    const float* __restrict__ charges,
    const float* __restrict__ voltage,
    const int*   __restrict__ mol,
    float*       __restrict__ out_mol,
    int n_atoms)
{
  int i = blockIdx.x * blockDim.x + threadIdx.x;
  if (i < n_atoms) {
    float e = 0.5f * voltage[i] * charges[i];
    atomicAdd(&out_mol[mol[i]], e);
  }
}

extern "C" void kernel_launch(void* const* d_in, const int* in_sizes, int n_in,
                              void* d_out, int out_size, void* d_ws, size_t ws_size,
                              hipStream_t stream) {
  const float* charges     = (const float*)d_in[0];
  const float* pair_dist   = (const float*)d_in[1];
  const int*   pair_first  = (const int*)d_in[2];
  const int*   pair_second = (const int*)d_in[3];
  const int*   mol_index   = (const int*)d_in[4];
  // d_in[5] = n_molecules (device int scalar, layout derived from sizes instead)
  const float* ecf         = (const float*)d_in[6];

  const int n_atoms = in_sizes[0];   // charges flat count (N_ATOMS x 1)
  const int n_pairs = in_sizes[1];
  const int n_mol   = out_size - n_atoms;

  float* out_mol  = (float*)d_out;          // first n_mol floats
  float* out_volt = (float*)d_out + n_mol;  // then n_atoms floats

  // 1) zero outputs (accumulators)
  coulomb_zero_f32<<<(out_size + 255) / 256, 256, 0, stream>>>((float*)d_out, out_size);

  // 2) pair scatter-add -> voltage_atom (persistent blocks, async pipeline)
  coulomb_pair_voltage_kernel<<<4096, 256, 0, stream>>>(
      charges, pair_dist, pair_first, pair_second, ecf, out_volt, n_pairs);

  // 3) per-atom energy -> molecule scatter-add
  coulomb_atom_molecule_kernel<<<(n_atoms + 255) / 256, 256, 0, stream>>>(
      charges, out_volt, mol_index, out_mol, n_atoms);
}